// SPAIRGlimpseEncoder_64269890617421
// MI455X (gfx1250) — compile-verified
//
#include <hip/hip_runtime.h>
#include <math.h>

// ---------------------------------------------------------------------------
// SPAIR glimpse encoder for MI455X (gfx1250, wave32).
// All matmuls use V_WMMA_F32_16X16X4_F32 (exact fp32, native CDNA5 tile).
// Segment-max uses u32 atomicMax on float bits (valid: ReLU output >= 0).
// Conv/GEMM kernels are template-specialized so K/N loops fully unroll:
// constant-folded pad checks, b64-merged operand loads, back-to-back WMMAs.
// ---------------------------------------------------------------------------

typedef __attribute__((ext_vector_type(2))) float v2f;
typedef __attribute__((ext_vector_type(8))) float v8f;

__device__ __forceinline__ float celu1(float x) {
    return (x > 0.0f) ? x : expm1f(x);
}
__device__ __forceinline__ float softplus1(float x) {
    return (x > 20.0f) ? x : log1pf(__expf(x));
}

// ---------------------------------------------------------------------------
__global__ void k_zero(float* __restrict__ p, long n) {
    long i = (long)blockIdx.x * blockDim.x + threadIdx.x;
    long stride = (long)gridDim.x * blockDim.x;
    for (; i < n; i += stride) p[i] = 0.0f;
}

// scatter-add of 3-vectors + count (for segment mean)
__global__ void k_scatter_mean(const float* __restrict__ pos,
                               const int* __restrict__ idx,
                               float* __restrict__ sum, float* __restrict__ cnt,
                               int n) {
    int i = blockIdx.x * blockDim.x + threadIdx.x;
    if (i >= n) return;
    int c = idx[i];
    atomicAdd(&sum[3 * c + 0], pos[3 * i + 0]);
    atomicAdd(&sum[3 * c + 1], pos[3 * i + 1]);
    atomicAdd(&sum[3 * c + 2], pos[3 * i + 2]);
    atomicAdd(&cnt[c], 1.0f);
}

__global__ void k_finalize_mean(float* __restrict__ sum,
                                const float* __restrict__ cnt, int m) {
    int i = blockIdx.x * blockDim.x + threadIdx.x;
    if (i >= m) return;
    float inv = 1.0f / fmaxf(cnt[i], 1.0f);
    sum[3 * i + 0] *= inv;
    sum[3 * i + 1] *= inv;
    sum[3 * i + 2] *= inv;
}

// feature fetch for point-conv: [X(p, 0..XDIM-1), rel.x, rel.y, rel.z, 0-pad]
template <int XDIM, int KDIM>
__device__ __forceinline__ float pc_feat(const float* __restrict__ X, int p,
                                         int k, float rx, float ry, float rz) {
    if (k >= KDIM) return 0.0f;
    if (k < XDIM) return X[(size_t)p * XDIM + k];
    int r = k - XDIM;
    return (r == 0) ? rx : ((r == 1) ? ry : rz);
}

// ---------------------------------------------------------------------------
// PointConv message pass:  msg = relu([X, pos_in - pos_out[idx]] @ W + bias)
// then segment-max scatter into agg[idx][:] (u32 atomicMax on float bits).
// One wave = one 16-row tile. kc-outer / nt-inner with NT accumulators so
// each A chunk is fetched once and feeds all column tiles.
// A layout: lane<16 holds K=4kc+{0,1} of point base+lane; lane>=16 holds
// K=4kc+{2,3} of point base+(lane-16). B mirrored. D: VGPR j -> row j / j+8.
// ---------------------------------------------------------------------------
template <int XDIM, int NC>
__global__ void k_point_conv_wmma(const float* __restrict__ X,
                                  const float* __restrict__ pos_in,
                                  const float* __restrict__ pos_out,  // null => zeros
                                  const int* __restrict__ idx,
                                  const float* __restrict__ W,    // [KDIM, NC] row-major
                                  const float* __restrict__ bias, // [NC]
                                  float* __restrict__ agg,        // [M, NC], zero-init
                                  int E) {
    constexpr int KDIM = XDIM + 3;
    constexpr int KCH  = (KDIM + 3) / 4;
    constexpr int NT   = NC / 16;

    const int lane = threadIdx.x & 31;
    const int wave = threadIdx.x >> 5;
    const int tile = blockIdx.x * (blockDim.x >> 5) + wave;
    if (tile >= (E + 15) >> 4) return;      // wave-uniform: EXEC stays all-1s

    const int base = tile << 4;
    const int half = lane >> 4;             // 0 or 1
    const int n    = lane & 15;
    const int kb   = half << 1;             // K sub-offset: 0 or 2

    // this lane's A-row point (lanes l and l+16 share point base+n)
    int pa = base + n; if (pa >= E) pa = E - 1;
    int ca = idx[pa];
    float rx = pos_in[3 * pa + 0];
    float ry = pos_in[3 * pa + 1];
    float rz = pos_in[3 * pa + 2];
    if (pos_out) {
        rx -= pos_out[3 * ca + 0];
        ry -= pos_out[3 * ca + 1];
        rz -= pos_out[3 * ca + 2];
    }

    v8f acc[NT];
#pragma unroll
    for (int nt = 0; nt < NT; ++nt) acc[nt] = (v8f){0.f,0.f,0.f,0.f,0.f,0.f,0.f,0.f};

#pragma unroll
    for (int kc = 0; kc < KCH; ++kc) {
        const int k0 = (kc << 2) + kb;
        v2f a;
        a.x = pc_feat<XDIM, KDIM>(X, pa, k0,     rx, ry, rz);
        a.y = pc_feat<XDIM, KDIM>(X, pa, k0 + 1, rx, ry, rz);
#pragma unroll
        for (int nt = 0; nt < NT; ++nt) {
            const int ncol = (nt << 4) + n;
            v2f b;
            b.x = (k0     < KDIM) ? W[(size_t)k0       * NC + ncol] : 0.0f;
            b.y = (k0 + 1 < KDIM) ? W[(size_t)(k0 + 1) * NC + ncol] : 0.0f;
            acc[nt] = __builtin_amdgcn_wmma_f32_16x16x4_f32(
                false, a, false, b, (short)0, acc[nt], false, false);
        }
    }

    float bv[NT];
#pragma unroll
    for (int nt = 0; nt < NT; ++nt) bv[nt] = bias[(nt << 4) + n];

#pragma unroll
    for (int j = 0; j < 8; ++j) {
        int row = base + j + (half << 3);
        bool ok = row < E;
        int c = idx[ok ? row : (E - 1)];
#pragma unroll
        for (int nt = 0; nt < NT; ++nt) {
            if (ok) {
                float v = fmaxf(acc[nt][j] + bv[nt], 0.0f);  // relu => v >= 0
                atomicMax((unsigned int*)&agg[(size_t)c * NC + ((nt << 4) + n)],
                          __float_as_uint(v));
            }
        }
    }
}

// ---------------------------------------------------------------------------
// Dense GEMM + bias + activation: Y[rows,NC] = act(X[rows,K] @ W[K,NC] + b)
// act: 0 = celu, 1 = identity. grid.y indexes 16-wide column tiles.
// ---------------------------------------------------------------------------
template <int K, int NC>
__global__ void k_gemm_wmma(const float* __restrict__ X,
                            const float* __restrict__ W,
                            const float* __restrict__ bias,
                            float* __restrict__ Y,
                            int rows, int act) {
    const int lane = threadIdx.x & 31;
    const int wave = threadIdx.x >> 5;
    const int rowTile = blockIdx.x * (blockDim.x >> 5) + wave;
    if (rowTile >= (rows + 15) >> 4) return; // wave-uniform

    const int base = rowTile << 4;
    const int half = lane >> 4;
    const int n    = lane & 15;
    const int kb   = half << 1;
    const int ncol = (blockIdx.y << 4) + n;

    int ra = base + n; if (ra >= rows) ra = rows - 1;

    v8f acc = {0.f, 0.f, 0.f, 0.f, 0.f, 0.f, 0.f, 0.f};
#pragma unroll
    for (int kc = 0; kc < K; kc += 4) {   // K is a multiple of 4 for all uses
        const int k0 = kc + kb;
        v2f a, b;
        a.x = X[(size_t)ra * K + k0];
        a.y = X[(size_t)ra * K + k0 + 1];
        b.x = W[(size_t)k0       * NC + ncol];
        b.y = W[(size_t)(k0 + 1) * NC + ncol];
        acc = __builtin_amdgcn_wmma_f32_16x16x4_f32(
            false, a, false, b, (short)0, acc, false, false);
    }
    const float bv = bias[ncol];
#pragma unroll
    for (int j = 0; j < 8; ++j) {
        int row = base + j + (half << 3);
        if (row < rows) {
            float v = acc[j] + bv;
            if (act == 0) v = celu1(v);
            Y[(size_t)row * NC + ncol] = v;
        }
    }
}

// ---------------------------------------------------------------------------
// Head: split lin[G,256] -> mu/sigma, z = mu + softplus(s)*eps, write outputs.
// d_out layout: z_what[G*64] | z_mask[G*64] | mu[G*128] | sigma[G*128] | f3[G*256]
// ---------------------------------------------------------------------------
__global__ void k_head(const float* __restrict__ lin,
                       const float* __restrict__ eps,
                       float* __restrict__ out, int G) {
    int i = blockIdx.x * blockDim.x + threadIdx.x;
    if (i >= G * 128) return;
    int g = i >> 7, ch = i & 127;
    float mu = lin[(size_t)g * 256 + ch];
    float s  = lin[(size_t)g * 256 + 128 + ch];
    float sigma = softplus1(s);
    float z = mu + sigma * eps[i];
    out[(size_t)G * 128 + i] = mu;
    out[(size_t)G * 256 + i] = sigma;
    if (ch < 64) out[(size_t)g * 64 + ch] = z;
    else         out[(size_t)G * 64 + (size_t)g * 64 + (ch - 64)] = z;
}

// ---------------------------------------------------------------------------
extern "C" void kernel_launch(void* const* d_in, const int* in_sizes, int n_in,
                              void* d_out, int out_size, void* d_ws, size_t ws_size,
                              hipStream_t stream) {
    const float* rgb  = (const float*)d_in[0];
    const float* pos  = (const float*)d_in[1];
    const float* W1l  = (const float*)d_in[3];
    const float* b1l  = (const float*)d_in[4];
    const float* W1g  = (const float*)d_in[5];
    const float* b1g  = (const float*)d_in[6];
    const float* W2l  = (const float*)d_in[7];
    const float* b2l  = (const float*)d_in[8];
    const float* W2g  = (const float*)d_in[9];
    const float* b2g  = (const float*)d_in[10];
    const float* W3l  = (const float*)d_in[11];
    const float* b3l  = (const float*)d_in[12];
    const float* W3g  = (const float*)d_in[13];
    const float* b3g  = (const float*)d_in[14];
    const float* Wlin = (const float*)d_in[15];
    const float* blin = (const float*)d_in[16];
    const float* eps  = (const float*)d_in[17];
    const int* cluster1 = (const int*)d_in[19];
    const int* cluster2 = (const int*)d_in[20];
    const int* batch2   = (const int*)d_in[21];

    const int N  = in_sizes[0];          // points
    const int G  = in_sizes[17] / 128;   // glimpses
    const int M1 = in_sizes[20];         // len(cluster2) == num_voxels1
    const int M2 = in_sizes[21];         // len(batch2)   == num_voxels2

    // workspace layout: accumulators (zeroed) first, then scratch tensors
    float* ws   = (float*)d_ws;
    float* sum1 = ws;                          // 3*M1 -> becomes ps1
    float* cnt1 = sum1 + (size_t)3 * M1;       // M1
    float* agg1 = cnt1 + (size_t)M1;           // 16*M1
    float* sum2 = agg1 + (size_t)16 * M1;      // 3*M2 -> becomes ps2
    float* cnt2 = sum2 + (size_t)3 * M2;       // M2
    float* agg2 = cnt2 + (size_t)M2;           // 64*M2
    float* agg3 = agg2 + (size_t)64 * M2;      // 128*G
    float* f1   = agg3 + (size_t)128 * G;      // 32*M1  (fully overwritten)
    float* f2   = f1   + (size_t)32 * M1;      // 128*M2 (fully overwritten)
    float* lin  = f2   + (size_t)128 * M2;     // 256*G  (fully overwritten)

    float* out = (float*)d_out;
    float* f3  = out + (size_t)384 * G;        // f3 slot of d_out, 256*G

    const long zeroN = (long)20 * M1 + (long)68 * M2 + (long)128 * G;
    k_zero<<<2048, 256, 0, stream>>>(ws, zeroN);

    // ---- stage 1: voxel mean (0.25) + PointConv [rgb,rel](4) -> 16 -> 32
    k_scatter_mean<<<(N + 255) / 256, 256, 0, stream>>>(pos, cluster1, sum1, cnt1, N);
    k_finalize_mean<<<(M1 + 255) / 256, 256, 0, stream>>>(sum1, cnt1, M1);
    {
        int tiles = (N + 15) >> 4;
        k_point_conv_wmma<1, 16><<<(tiles + 3) / 4, 128, 0, stream>>>(
            rgb, pos, sum1, cluster1, W1l, b1l, agg1, N);
        int rt = (M1 + 15) >> 4;
        dim3 g((rt + 3) / 4, 32 / 16);
        k_gemm_wmma<16, 32><<<g, 128, 0, stream>>>(agg1, W1g, b1g, f1, M1, 0);
    }

    // ---- stage 2: voxel mean (0.5) + PointConv [f1,rel](35) -> 64 -> 128
    k_scatter_mean<<<(M1 + 255) / 256, 256, 0, stream>>>(sum1, cluster2, sum2, cnt2, M1);
    k_finalize_mean<<<(M2 + 255) / 256, 256, 0, stream>>>(sum2, cnt2, M2);
    {
        int tiles = (M1 + 15) >> 4;
        k_point_conv_wmma<32, 64><<<(tiles + 3) / 4, 128, 0, stream>>>(
            f1, sum1, sum2, cluster2, W2l, b2l, agg2, M1);
        int rt = (M2 + 15) >> 4;
        dim3 g((rt + 3) / 4, 128 / 16);
        k_gemm_wmma<64, 128><<<g, 128, 0, stream>>>(agg2, W2g, b2g, f2, M2, 0);
    }

    // ---- stage 3: pool to glimpse center (pos_out = 0): [f2,ps2](131)->128->256
    {
        int tiles = (M2 + 15) >> 4;
        k_point_conv_wmma<128, 128><<<(tiles + 3) / 4, 128, 0, stream>>>(
            f2, sum2, nullptr, batch2, W3l, b3l, agg3, M2);
        int rt = (G + 15) >> 4;
        dim3 g((rt + 3) / 4, 256 / 16);
        k_gemm_wmma<128, 256><<<g, 128, 0, stream>>>(agg3, W3g, b3g, f3, G, 0);
    }

    // ---- head: Linear(256,256) + reparameterized sample
    {
        int rt = (G + 15) >> 4;
        dim3 g((rt + 3) / 4, 256 / 16);
        k_gemm_wmma<256, 256><<<g, 128, 0, stream>>>(f3, Wlin, blin, lin, G, 1);
    }
    k_head<<<(G * 128 + 255) / 256, 256, 0, stream>>>(lin, eps, out, G);
}